// SAGE_79620103733916
// MI455X (gfx1250) — compile-verified
//
#include <hip/hip_runtime.h>

#define IN_DIM  128
#define HID_DIM 128
#define N_CLS   40

typedef __attribute__((ext_vector_type(2))) float v2f;
typedef __attribute__((ext_vector_type(8))) float v8f;

// D = A(16x4 f32) * B(4x16 f32) + C(16x16 f32), exact fp32 matrix op on CDNA5.
static __device__ __forceinline__ v8f wmma_f32x4(v2f a, v2f b, v8f c) {
  return __builtin_amdgcn_wmma_f32_16x16x4_f32(
      /*neg_a=*/false, a, /*neg_b=*/false, b,
      /*c_mod=*/(short)0, c, /*reuse_a=*/false, /*reuse_b=*/false);
}

// Hardware f32 atomic add (global_atomic_add_f32, no CAS loop). agg/deg live
// in coarse-grained device workspace, so the unsafe (hw) path is correct.
static __device__ __forceinline__ void atomic_add_f32(float* p, float v) {
#if defined(__HIP_DEVICE_COMPILE__)
  unsafeAtomicAdd(p, v);
#else
  atomicAdd(p, v);
#endif
}

// ---------------------------------------------------------------------------
// Zero-fill f32 buffer
// ---------------------------------------------------------------------------
__global__ __launch_bounds__(256) void zero_f32(float* __restrict__ p, size_t n) {
  size_t i = (size_t)blockIdx.x * 256u + threadIdx.x;
  if (i < n) p[i] = 0.0f;
}

// ---------------------------------------------------------------------------
// Edge scatter: agg[dst] += feat[src] (128 f32 per edge), optional degree.
// One wave (32 lanes) per edge; each lane moves 4 contiguous floats via a
// b128 load + 4 hardware global_atomic_add_f32.
// ---------------------------------------------------------------------------
__global__ __launch_bounds__(256) void scatter_add(
    const int* __restrict__ src, const int* __restrict__ dst,
    const float* __restrict__ feat, float* __restrict__ agg,
    float* __restrict__ deg, int nEdges, int addDeg) {
  int e = blockIdx.x * 8 + (threadIdx.x >> 5);
  if (e >= nEdges) return;
  int lane = threadIdx.x & 31;
  int s = src[e];
  int d = dst[e];
  const float4 v = ((const float4*)(feat + (size_t)s * IN_DIM))[lane];
  float* a = agg + (size_t)d * IN_DIM + lane * 4;
  atomic_add_f32(a + 0, v.x);
  atomic_add_f32(a + 1, v.y);
  atomic_add_f32(a + 2, v.z);
  atomic_add_f32(a + 3, v.w);
  if (addDeg && lane == 0) atomic_add_f32(deg + d, 1.0f);
}

// ---------------------------------------------------------------------------
// Layer 1: h = relu( (agg/deg) @ Wl1^T + bl1 + x @ Wr1^T ), then L2-normalize
// rows. Block = 256 threads = 8 waves; block owns 16 nodes (M), wave w owns
// output features [16w, 16w+16) (N). K = 128 in steps of 4 (f32 WMMA).
//
// f32 WMMA fragment layouts (ISA 7.12.2):
//   A 16x4 : VGPR0 = K=0 (lanes 0-15) / K=2 (lanes 16-31); VGPR1 = K=1 / K=3
//   B 4x16 : VGPR0 = K=0 row / K=2 row;                   VGPR1 = K=1 / K=3
//   C/D    : VGPR r = row M=r (lanes 0-15) / M=r+8 (lanes 16-31), N = lane&15
// ---------------------------------------------------------------------------
__global__ __launch_bounds__(256) void gemm1_relu_norm(
    const float* __restrict__ agg, const float* __restrict__ deg,
    const float* __restrict__ x, const float* __restrict__ Wl,
    const float* __restrict__ bl, const float* __restrict__ Wr,
    float* __restrict__ h) {
  __shared__ float hs[16][HID_DIM];   // relu'd tile, 8 KB
  __shared__ float psum[16][16];      // per-row partial square sums
  __shared__ float rscale[16];        // per-row 1/max(norm, eps)

  const int node0 = blockIdx.x * 16;
  const int lane  = threadIdx.x & 31;
  const int wave  = threadIdx.x >> 5;     // 0..7 -> N tile
  const int half  = lane >> 4;            // 0 or 1
  const int l15   = lane & 15;
  const int node  = node0 + l15;          // A-matrix row for this lane
  const int n     = wave * 16 + l15;      // B-matrix column (output feature)
  const int koff  = half * 2;             // K sub-offset per half-wave

  const float dinv = 1.0f / fmaxf(deg[node], 1.0f);
  const float* aggRow = agg + (size_t)node * IN_DIM;
  const float* xRow   = x   + (size_t)node * IN_DIM;
  const float* wlRow  = Wl  + (size_t)n * IN_DIM;
  const float* wrRow  = Wr  + (size_t)n * IN_DIM;

  v8f accL = {};  // (agg/deg) @ Wl^T
  v8f accR = {};  // x @ Wr^T
#pragma unroll 4
  for (int k = 0; k < IN_DIM; k += 4) {
    const int kk = k + koff;
    const float2 ag = *(const float2*)(aggRow + kk);
    const float2 bw = *(const float2*)(wlRow + kk);
    v2f aA = { ag.x * dinv, ag.y * dinv };
    v2f bA = { bw.x, bw.y };
    accL = wmma_f32x4(aA, bA, accL);

    const float2 xx = *(const float2*)(xRow + kk);
    const float2 wr = *(const float2*)(wrRow + kk);
    v2f aB = { xx.x, xx.y };
    v2f bB = { wr.x, wr.y };
    accR = wmma_f32x4(aB, bB, accR);
  }

  const float bias = bl[n];
#pragma unroll
  for (int r = 0; r < 8; ++r) {
    float v = accL[r] + accR[r] + bias;
    v = fmaxf(v, 0.0f);                       // ReLU
    hs[r + half * 8][wave * 16 + l15] = v;
  }
  __syncthreads();

  // Deterministic row-wise sum of squares: 16 partials per row, tree style.
  {
    const int r = threadIdx.x >> 4;           // row 0..15
    const int c = (threadIdx.x & 15) * 8;     // 8-elem chunk
    float s = 0.0f;
#pragma unroll
    for (int i = 0; i < 8; ++i) { float v = hs[r][c + i]; s += v * v; }
    psum[r][threadIdx.x & 15] = s;
  }
  __syncthreads();
  if (threadIdx.x < 16) {
    float t = 0.0f;
#pragma unroll
    for (int j = 0; j < 16; ++j) t += psum[threadIdx.x][j];
    rscale[threadIdx.x] = 1.0f / fmaxf(sqrtf(t), 1e-12f);
  }
  __syncthreads();

  // Normalized writeback of h.
  {
    const int r = threadIdx.x >> 4;
    const int c = (threadIdx.x & 15) * 8;
    const float s = rscale[r];
    float* dstp = h + (size_t)(node0 + r) * HID_DIM + c;
#pragma unroll
    for (int i = 0; i < 8; ++i) dstp[i] = hs[r][c + i] * s;
  }
}

// ---------------------------------------------------------------------------
// Layer 2: out = (agg/deg) @ Wl2^T + bl2 + h @ Wr2^T  -> [N, 40]
// Block = 96 threads = 3 waves covering N = 0..47; writes masked to N < 40.
// Weight-row reads clamped to stay in bounds (garbage columns never stored).
// ---------------------------------------------------------------------------
__global__ __launch_bounds__(96) void gemm2(
    const float* __restrict__ agg, const float* __restrict__ deg,
    const float* __restrict__ h, const float* __restrict__ Wl,
    const float* __restrict__ bl, const float* __restrict__ Wr,
    float* __restrict__ out) {
  const int node0 = blockIdx.x * 16;
  const int lane  = threadIdx.x & 31;
  const int wave  = threadIdx.x >> 5;     // 0..2
  const int half  = lane >> 4;
  const int l15   = lane & 15;
  const int node  = node0 + l15;
  const int n     = wave * 16 + l15;      // 0..47
  const int nc    = (n < N_CLS) ? n : (N_CLS - 1);   // clamp for safe loads
  const int koff  = half * 2;

  const float dinv = 1.0f / fmaxf(deg[node], 1.0f);
  const float* aggRow = agg + (size_t)node * HID_DIM;
  const float* hRow   = h   + (size_t)node * HID_DIM;
  const float* wlRow  = Wl  + (size_t)nc * HID_DIM;
  const float* wrRow  = Wr  + (size_t)nc * HID_DIM;

  v8f accL = {};
  v8f accR = {};
#pragma unroll 4
  for (int k = 0; k < HID_DIM; k += 4) {
    const int kk = k + koff;
    const float2 ag = *(const float2*)(aggRow + kk);
    const float2 bw = *(const float2*)(wlRow + kk);
    v2f aA = { ag.x * dinv, ag.y * dinv };
    v2f bA = { bw.x, bw.y };
    accL = wmma_f32x4(aA, bA, accL);

    const float2 hh = *(const float2*)(hRow + kk);
    const float2 wr = *(const float2*)(wrRow + kk);
    v2f aB = { hh.x, hh.y };
    v2f bB = { wr.x, wr.y };
    accR = wmma_f32x4(aB, bB, accR);
  }

  const float bias = bl[nc];
  if (n < N_CLS) {
#pragma unroll
    for (int r = 0; r < 8; ++r) {
      const int row = r + half * 8;
      out[(size_t)(node0 + row) * N_CLS + n] = accL[r] + accR[r] + bias;
    }
  }
}

// ---------------------------------------------------------------------------
extern "C" void kernel_launch(void* const* d_in, const int* in_sizes, int n_in,
                              void* d_out, int out_size, void* d_ws, size_t ws_size,
                              hipStream_t stream) {
  const float* x   = (const float*)d_in[0];
  const int*   ei  = (const int*)d_in[1];     // edge_index [2, E] (int32)
  const float* Wl1 = (const float*)d_in[2];
  const float* bl1 = (const float*)d_in[3];
  const float* Wr1 = (const float*)d_in[4];
  const float* Wl2 = (const float*)d_in[5];
  const float* bl2 = (const float*)d_in[6];
  const float* Wr2 = (const float*)d_in[7];
  float* out = (float*)d_out;

  const int E = in_sizes[1] / 2;
  const int N = in_sizes[0] / IN_DIM;         // 100000 (multiple of 16)
  const int* src = ei;
  const int* dst = ei + E;

  // Workspace layout (floats): agg[N*128] | deg[N] | h[N*128]  (~103 MB)
  float* agg = (float*)d_ws;
  float* deg = agg + (size_t)N * IN_DIM;
  float* hbuf = deg + N;

  const int edgeBlocks = (E + 7) / 8;
  const int mTiles = N / 16;

  // Layer 1
  {
    size_t nz = (size_t)N * IN_DIM + (size_t)N;   // agg + deg contiguous
    zero_f32<<<dim3((unsigned)((nz + 255) / 256)), dim3(256), 0, stream>>>(agg, nz);
  }
  scatter_add<<<dim3(edgeBlocks), dim3(256), 0, stream>>>(src, dst, x, agg, deg, E, 1);
  gemm1_relu_norm<<<dim3(mTiles), dim3(256), 0, stream>>>(agg, deg, x, Wl1, bl1, Wr1, hbuf);

  // Layer 2
  {
    size_t nz = (size_t)N * IN_DIM;
    zero_f32<<<dim3((unsigned)((nz + 255) / 256)), dim3(256), 0, stream>>>(agg, nz);
  }
  scatter_add<<<dim3(edgeBlocks), dim3(256), 0, stream>>>(src, dst, hbuf, agg, deg, E, 0);
  gemm2<<<dim3(mTiles), dim3(96), 0, stream>>>(agg, deg, hbuf, Wl2, bl2, Wr2, out);
}